// akinciTensionModule_47021301957211
// MI455X (gfx1250) — compile-verified
//
#include <hip/hip_runtime.h>

#ifndef GAMMA_F
#define GAMMA_F 1.0f
#endif
#define SUPPORT_F 0.05f

// ---------------------------------------------------------------------------
// Zero-init normals workspace and output (must be re-done every call: the
// harness poisons buffers once and never restores them between replays).
// ---------------------------------------------------------------------------
__global__ void akinci_zero_kernel(float* __restrict__ normals,
                                   float* __restrict__ out, int n2) {
  int t = blockIdx.x * blockDim.x + threadIdx.x;
  if (t < n2) {
    normals[t] = 0.0f;
    out[t]     = 0.0f;
  }
}

// ---------------------------------------------------------------------------
// Phase 1: normals[i] += SUPPORT * (area[j]/dens[j]) * gradW(q) * dist
//   gradW factor: C*(-20*q*max(1-q,0)^3)/h, and SUPPORT/h == 1 cancels.
// ---------------------------------------------------------------------------
__global__ void akinci_normals_kernel(const int* __restrict__ ni,
                                      const int* __restrict__ nj,
                                      const float* __restrict__ areas,
                                      const float* __restrict__ dens,
                                      const float* __restrict__ q,
                                      const float2* __restrict__ dist,
                                      float* __restrict__ normals, int E) {
  int e = blockIdx.x * blockDim.x + threadIdx.x;
  if (e >= E) return;

  // gfx1250 speculative prefetch of the streaming edge arrays (dropped if OOB)
  __builtin_prefetch(&q[e + 16384], 0, 0);
  __builtin_prefetch(&dist[e + 8192], 0, 0);

  int i = ni[e];
  int j = nj[e];
  float qe = q[e];
  float2 d = dist[e];          // unit direction, coalesced global_load_b64

  const float C = 7.0f / (3.14159265358979323846f * SUPPORT_F * SUPPORT_F);
  float c = 1.0f - qe;
  c = fmaxf(c, 0.0f);
  float dw = C * (-20.0f * qe) * (c * c * c);   // SUPPORT/h cancelled
  float fac = areas[j] / dens[j];               // L2-resident gathers (400 KB)
  float s = fac * dw;

  atomicAdd(&normals[2 * i + 0], s * d.x);      // global_atomic_add_f32 in L2
  atomicAdd(&normals[2 * i + 1], s * d.y);
}

// ---------------------------------------------------------------------------
// Phase 2: out[i] += -GAMMA*(normals[i]-normals[j])
//                  + GAMMA*area[j]*restDens[j]*sel(q)*dist
//   sel(q) = (q<=0.5) ? 128*((q-1)q)^3 + 1 : 64*((q-1)q)^3
//   (cohesion's two minus signs cancel)
// ---------------------------------------------------------------------------
__global__ void akinci_forces_kernel(const int* __restrict__ ni,
                                     const int* __restrict__ nj,
                                     const float* __restrict__ areas,
                                     const float* __restrict__ restD,
                                     const float* __restrict__ q,
                                     const float2* __restrict__ dist,
                                     const float2* __restrict__ normals,
                                     float* __restrict__ out, int E) {
  int e = blockIdx.x * blockDim.x + threadIdx.x;
  if (e >= E) return;

  __builtin_prefetch(&q[e + 16384], 0, 0);
  __builtin_prefetch(&dist[e + 8192], 0, 0);

  int i = ni[e];
  int j = nj[e];
  float qe = q[e];
  float2 d  = dist[e];
  float2 Ni = normals[i];      // 800 KB array -> L2 hits, global_load_b64
  float2 Nj = normals[j];

  float t = (qe - 1.0f) * qe;
  float p = t * t * t;                               // (q-1)^3 * q^3
  float sel = (qe <= 0.5f) ? fmaf(128.0f, p, 1.0f) : 64.0f * p;
  float coh = areas[j] * restD[j] * sel * GAMMA_F;

  float cx = -GAMMA_F * (Ni.x - Nj.x) + coh * d.x;
  float cy = -GAMMA_F * (Ni.y - Nj.y) + coh * d.y;

  atomicAdd(&out[2 * i + 0], cx);
  atomicAdd(&out[2 * i + 1], cy);
}

// ---------------------------------------------------------------------------
// Launch: zero -> normals scatter -> force scatter (stream-ordered).
// d_in order: neighbors[2E] i32, areas[N], dens[N], restDens[N], q[E], dist[E,2]
// d_out: [N,2] f32.  d_ws: normals [N,2] f32 (800 KB).
// ---------------------------------------------------------------------------
extern "C" void kernel_launch(void* const* d_in, const int* in_sizes, int n_in,
                              void* d_out, int out_size, void* d_ws, size_t ws_size,
                              hipStream_t stream) {
  const int*   neigh = (const int*)  d_in[0];
  const float* areas = (const float*)d_in[1];
  const float* dens  = (const float*)d_in[2];
  const float* restD = (const float*)d_in[3];
  const float* q     = (const float*)d_in[4];
  const float2* dist = (const float2*)d_in[5];

  const int E = in_sizes[4];        // fluidRadialDistances is [E]
  const int N = in_sizes[1];        // fluidAreas is [N]
  const int* ni = neigh;            // neighbors[0,:]
  const int* nj = neigh + E;        // neighbors[1,:]

  float* out     = (float*)d_out;   // [N,2]
  float* normals = (float*)d_ws;    // [N,2] scratch

  const int BLK = 256;              // 8 wave32 per block
  int n2 = 2 * N;

  akinci_zero_kernel<<<(n2 + BLK - 1) / BLK, BLK, 0, stream>>>(normals, out, n2);

  int eb = (E + BLK - 1) / BLK;
  akinci_normals_kernel<<<eb, BLK, 0, stream>>>(ni, nj, areas, dens, q, dist,
                                                normals, E);
  akinci_forces_kernel<<<eb, BLK, 0, stream>>>(ni, nj, areas, restD, q, dist,
                                               (const float2*)normals, out, E);
}